// EthicalRelationReasoning_42348377538831
// MI455X (gfx1250) — compile-verified
//
#include <hip/hip_runtime.h>
#include <math.h>

#define NN 50000
#define EE 600000
#define GG 50
#define ETOT (EE + NN)

typedef __attribute__((ext_vector_type(2))) float v2f;
typedef __attribute__((ext_vector_type(8))) float v8f;

// ---------------- utility ----------------

__global__ void fill_kernel(float* __restrict__ p, float v, int n) {
    int t = blockIdx.x * blockDim.x + threadIdx.x;
    if (t < n) p[t] = v;
}

__device__ __forceinline__ float atomicMaxF(float* addr, float v) {
    // sign-aware int-bits max trick (monotone mapping of IEEE floats)
    if (v >= 0.0f)
        return __int_as_float(atomicMax((int*)addr, __float_as_int(v)));
    else
        return __uint_as_float(atomicMin((unsigned int*)addr, __float_as_uint(v)));
}

__device__ __forceinline__ void edge_sd(const int* __restrict__ ei, int e,
                                        int& s, int& d) {
    if (e < EE) { s = ei[e]; d = ei[EE + e]; }
    else        { s = e - EE; d = e - EE; }   // self loop
}

// ---------------- encoder: h = x @ enc_w + enc_b  ([N,5]x[5,64]) ----------------

__global__ void encoder_kernel(const float* __restrict__ x,
                               const float* __restrict__ W,
                               const float* __restrict__ b,
                               float* __restrict__ h) {
    int t = blockIdx.x * blockDim.x + threadIdx.x;
    if (t >= NN * 64) return;
    int n = t >> 6, c = t & 63;
    float acc = b[c];
#pragma unroll
    for (int k = 0; k < 5; ++k) acc += x[n * 5 + k] * W[k * 64 + c];
    h[t] = acc;
}

// ---------------- fp32 WMMA GEMM: C[N,COLS] = A[N,DIN] @ B[DIN,COLS] ----------------
// One 16x16 output tile per wave32, K stepped by 4 via V_WMMA_F32_16X16X4_F32.

template <int DIN, int COLS>
__global__ void gemm_wmma_f32(const float* __restrict__ A,
                              const float* __restrict__ B,
                              float* __restrict__ C, int ntilesM) {
    const int NT = COLS / 16;
    int wave = (blockIdx.x * blockDim.x + threadIdx.x) >> 5;  // wave-uniform
    if (wave >= ntilesM * NT) return;                          // whole-wave exit
    int tm = wave / NT, tn = wave % NT;
    int lane = threadIdx.x & 31;
    int half = lane >> 4;        // 0: K={0,1}/M rows 0-7 of D, 1: K={2,3}/M rows 8-15
    int l16 = lane & 15;

    const float* arow = A + (size_t)(tm * 16 + l16) * DIN;  // A row for M=l16
    const float* bcol = B + tn * 16 + l16;                  // B column N=l16

    v8f acc = {};
#pragma unroll
    for (int k = 0; k < DIN; k += 4) {
        int ka = k + half * 2;
        v2f a, b;
        a.x = arow[ka];
        a.y = arow[ka + 1];
        b.x = bcol[(size_t)ka * COLS];
        b.y = bcol[(size_t)(ka + 1) * COLS];
        acc = __builtin_amdgcn_wmma_f32_16x16x4_f32(
            /*neg_a=*/false, a, /*neg_b=*/false, b,
            /*c_mod=*/(short)0, acc, /*reuse_a=*/false, /*reuse_b=*/false);
    }
    // D layout: VGPR r -> M = r + half*8, N = l16
    float* crow = C + (size_t)(tm * 16 + half * 8) * COLS + tn * 16 + l16;
#pragma unroll
    for (int r = 0; r < 8; ++r) crow[(size_t)r * COLS] = acc[r];
}

// ---------------- attention scores: es/ed [N,4] ----------------

template <int DOUT, int COLS>
__global__ void attn_score_kernel(const float* __restrict__ hW,
                                  const float* __restrict__ a_s,
                                  const float* __restrict__ a_d,
                                  float* __restrict__ es,
                                  float* __restrict__ ed) {
    int t = blockIdx.x * blockDim.x + threadIdx.x;
    if (t >= NN * 4) return;
    int n = t >> 2, hd = t & 3;
    const float* row = hW + (size_t)n * COLS + hd * DOUT;
    const float* va = a_s + hd * DOUT;
    const float* vd = a_d + hd * DOUT;
    float s1 = 0.f, s2 = 0.f;
    for (int c = 0; c < DOUT; ++c) {
        float h = row[c];
        s1 += h * va[c];
        s2 += h * vd[c];
    }
    es[t] = s1;
    ed[t] = s2;
}

// ---------------- edge softmax (3 passes) ----------------

__device__ __forceinline__ float edge_logit(const float* es, const float* ed,
                                            int s, int d, int hd) {
    float v = es[s * 4 + hd] + ed[d * 4 + hd];
    return v > 0.f ? v : 0.2f * v;  // leaky_relu(0.2)
}

__global__ void edge_max_kernel(const int* __restrict__ ei,
                                const float* __restrict__ es,
                                const float* __restrict__ ed,
                                float* __restrict__ m) {
    int e = blockIdx.x * blockDim.x + threadIdx.x;
    if (e >= ETOT) return;
    int s, d;
    edge_sd(ei, e, s, d);
#pragma unroll
    for (int hd = 0; hd < 4; ++hd)
        atomicMaxF(&m[d * 4 + hd], edge_logit(es, ed, s, d, hd));
}

__global__ void edge_sum_kernel(const int* __restrict__ ei,
                                const float* __restrict__ es,
                                const float* __restrict__ ed,
                                const float* __restrict__ m,
                                float* __restrict__ ssum) {
    int e = blockIdx.x * blockDim.x + threadIdx.x;
    if (e >= ETOT) return;
    int s, d;
    edge_sd(ei, e, s, d);
#pragma unroll
    for (int hd = 0; hd < 4; ++hd) {
        float ex = expf(edge_logit(es, ed, s, d, hd) - m[d * 4 + hd]);
        atomicAdd(&ssum[d * 4 + hd], ex);
    }
}

__global__ void edge_alpha_kernel(const int* __restrict__ ei,
                                  const float* __restrict__ es,
                                  const float* __restrict__ ed,
                                  const float* __restrict__ m,
                                  const float* __restrict__ ssum,
                                  float* __restrict__ alpha) {
    int t = blockIdx.x * blockDim.x + threadIdx.x;
    if (t >= ETOT * 4) return;
    int e = t >> 2, hd = t & 3;
    int s, d;
    edge_sd(ei, e, s, d);
    float ex = expf(edge_logit(es, ed, s, d, hd) - m[d * 4 + hd]);
    alpha[t] = ex / (ssum[d * 4 + hd] + 1e-16f);
}

// ---------------- aggregation with head-mean folded in ----------------

template <int DOUT, int COLS>
__global__ void edge_agg_kernel(const int* __restrict__ ei,
                                const float* __restrict__ hW,
                                const float* __restrict__ alpha,
                                float* __restrict__ agg) {
    int t = blockIdx.x * blockDim.x + threadIdx.x;
    if (t >= ETOT * DOUT) return;
    int e = t / DOUT, c = t % DOUT;
    int s, d;
    edge_sd(ei, e, s, d);
    const float* row = hW + (size_t)s * COLS;
    const float* al = alpha + e * 4;
    float val = al[0] * row[c] + al[1] * row[DOUT + c] +
                al[2] * row[2 * DOUT + c] + al[3] * row[3 * DOUT + c];
    atomicAdd(&agg[(size_t)d * DOUT + c], 0.25f * val);  // mean over 4 heads
}

// ---------------- BatchNorm over node axis ----------------

template <int DOUT>
__global__ void bn_stats_kernel(const float* __restrict__ agg,
                                const float* __restrict__ bias,
                                float* __restrict__ sum,
                                float* __restrict__ sumsq, int rowsPerBlock) {
    int c = threadIdx.x;  // blockDim == DOUT
    int r0 = blockIdx.x * rowsPerBlock;
    int r1 = r0 + rowsPerBlock;
    if (r1 > NN) r1 = NN;
    float b = bias[c];
    float s = 0.f, ss = 0.f;
    for (int r = r0; r < r1; ++r) {
        float v = agg[(size_t)r * DOUT + c] + b;
        s += v;
        ss += v * v;
    }
    atomicAdd(&sum[c], s);
    atomicAdd(&sumsq[c], ss);
}

template <int DOUT, bool RELU>
__global__ void bn_norm_kernel(const float* __restrict__ agg,
                               const float* __restrict__ bias,
                               const float* __restrict__ gamma,
                               const float* __restrict__ beta,
                               const float* __restrict__ sum,
                               const float* __restrict__ sumsq,
                               float* __restrict__ outp) {
    int t = blockIdx.x * blockDim.x + threadIdx.x;
    if (t >= NN * DOUT) return;
    int c = t % DOUT;
    const float invN = 1.0f / (float)NN;
    float mu = sum[c] * invN;
    float var = sumsq[c] * invN - mu * mu;
    float v = agg[t] + bias[c];
    float y = gamma[c] * (v - mu) * rsqrtf(var + 1e-5f) + beta[c];
    if (RELU) y = fmaxf(y, 0.f);
    outp[t] = y;
}

// ---------------- graph pooling ----------------

__global__ void pool_kernel(const float* __restrict__ h,
                            const int* __restrict__ batch,
                            float* __restrict__ counts,
                            float* __restrict__ gsum,
                            float* __restrict__ gmax) {
    int t = blockIdx.x * blockDim.x + threadIdx.x;
    if (t >= NN * 32) return;
    int n = t >> 5, c = t & 31;
    int g = batch[n];
    float v = h[t];
    atomicAdd(&gsum[g * 32 + c], v);
    atomicMaxF(&gmax[g * 32 + c], v);
    if (c == 0) atomicAdd(&counts[g], 1.0f);
}

__global__ void gemb_kernel(const float* __restrict__ counts,
                            const float* __restrict__ gsum,
                            const float* __restrict__ gmax,
                            float* __restrict__ gemb_ws,
                            float* __restrict__ gemb_out) {
    int t = blockIdx.x * blockDim.x + threadIdx.x;
    if (t >= GG * 32) return;
    int g = t >> 5;
    float cnt = counts[g];
    float s = gsum[t];
    float mean = s / fmaxf(cnt, 1.0f);
    float mx = (cnt > 0.f) ? gmax[t] : 0.f;
    float v = (mean + mx + s) * (1.0f / 3.0f);
    gemb_ws[t] = v;
    gemb_out[t] = v;
}

// ---------------- MLP heads ----------------

__global__ void heads_kernel(const float* __restrict__ gemb,
                             const float* __restrict__ ew1, const float* __restrict__ eb1,
                             const float* __restrict__ ew2, const float* __restrict__ eb2,
                             const float* __restrict__ mw1, const float* __restrict__ mb1,
                             const float* __restrict__ mw2, const float* __restrict__ mb2,
                             float* __restrict__ eth_out, float* __restrict__ man_out) {
    int t = threadIdx.x;      // blockDim == 128
    int which = t >> 6;       // 0 -> ethics, 1 -> manip
    int g = t & 63;
    if (g >= GG || which > 1) return;
    const float* w1 = which ? mw1 : ew1;
    const float* b1 = which ? mb1 : eb1;
    const float* w2 = which ? mw2 : ew2;
    const float* b2 = which ? mb2 : eb2;
    const float* gr = gemb + g * 32;
    float acc = b2[0];
#pragma unroll
    for (int j = 0; j < 16; ++j) {
        float hj = b1[j];
        for (int k = 0; k < 32; ++k) hj += gr[k] * w1[k * 16 + j];
        hj = fmaxf(hj, 0.f);
        acc += hj * w2[j];
    }
    float r = 1.0f / (1.0f + expf(-acc));
    if (which) man_out[g] = r; else eth_out[g] = r;
}

// ---------------- launcher ----------------

static inline int cdiv(int a, int b) { return (a + b - 1) / b; }

extern "C" void kernel_launch(void* const* d_in, const int* in_sizes, int n_in,
                              void* d_out, int out_size, void* d_ws, size_t ws_size,
                              hipStream_t stream) {
    (void)in_sizes; (void)n_in; (void)out_size; (void)ws_size;

    const float* x     = (const float*)d_in[0];
    const int*   ei    = (const int*)d_in[1];
    const int*   batch = (const int*)d_in[2];
    const float* enc_w = (const float*)d_in[3];
    const float* enc_b = (const float*)d_in[4];
    const float* W[3]  = {(const float*)d_in[5],  (const float*)d_in[11], (const float*)d_in[17]};
    const float* AS[3] = {(const float*)d_in[6],  (const float*)d_in[12], (const float*)d_in[18]};
    const float* AD[3] = {(const float*)d_in[7],  (const float*)d_in[13], (const float*)d_in[19]};
    const float* BI[3] = {(const float*)d_in[8],  (const float*)d_in[14], (const float*)d_in[20]};
    const float* GA[3] = {(const float*)d_in[9],  (const float*)d_in[15], (const float*)d_in[21]};
    const float* BE[3] = {(const float*)d_in[10], (const float*)d_in[16], (const float*)d_in[22]};
    const float* ew1 = (const float*)d_in[23];
    const float* eb1 = (const float*)d_in[24];
    const float* ew2 = (const float*)d_in[25];
    const float* eb2 = (const float*)d_in[26];
    const float* mw1 = (const float*)d_in[27];
    const float* mb1 = (const float*)d_in[28];
    const float* mw2 = (const float*)d_in[29];
    const float* mb2 = (const float*)d_in[30];

    // workspace layout (floats)
    float* ws = (float*)d_ws;
    size_t off = 0;
    float* hA    = ws + off; off += (size_t)NN * 64;    // layer input
    float* hW    = ws + off; off += (size_t)NN * 256;   // GEMM output
    float* esb   = ws + off; off += (size_t)NN * 4;
    float* edb   = ws + off; off += (size_t)NN * 4;
    float* mb    = ws + off; off += (size_t)NN * 4;
    float* sb    = ws + off; off += (size_t)NN * 4;
    float* alpha = ws + off; off += (size_t)ETOT * 4;
    float* agg   = ws + off; off += (size_t)NN * 64;
    float* stat  = ws + off; off += 128;                // [sum(64) | sumsq(64)]
    float* cnts  = ws + off; off += 64;
    float* gsum  = ws + off; off += GG * 32;
    float* gmaxb = ws + off; off += GG * 32;
    float* gembW = ws + off; off += GG * 32;

    float* out     = (float*)d_out;
    float* hOut    = out;                       // [N,32]
    float* gembOut = out + (size_t)NN * 32;     // [G,32]
    float* ethOut  = gembOut + GG * 32;         // [G]
    float* manOut  = ethOut + GG;               // [G]

    const float NEG_INF = -__builtin_inff();
    const int TB = 256;

    // encoder
    encoder_kernel<<<cdiv(NN * 64, TB), TB, 0, stream>>>(x, enc_w, enc_b, hA);

    for (int l = 0; l < 3; ++l) {
        const int DOUT = (l == 2) ? 32 : 64;
        const int COLS = DOUT * 4;
        const int ntilesM = NN / 16;                 // 3125
        const int ntiles = ntilesM * (COLS / 16);
        const int gemmBlocks = cdiv(ntiles, 8);      // 8 waves / 256-thread block

        if (l == 2) {
            gemm_wmma_f32<64, 128><<<gemmBlocks, TB, 0, stream>>>(hA, W[l], hW, ntilesM);
            attn_score_kernel<32, 128><<<cdiv(NN * 4, TB), TB, 0, stream>>>(hW, AS[l], AD[l], esb, edb);
        } else {
            gemm_wmma_f32<64, 256><<<gemmBlocks, TB, 0, stream>>>(hA, W[l], hW, ntilesM);
            attn_score_kernel<64, 256><<<cdiv(NN * 4, TB), TB, 0, stream>>>(hW, AS[l], AD[l], esb, edb);
        }

        fill_kernel<<<cdiv(NN * 4, TB), TB, 0, stream>>>(mb, NEG_INF, NN * 4);
        fill_kernel<<<cdiv(NN * 4, TB), TB, 0, stream>>>(sb, 0.f, NN * 4);
        edge_max_kernel<<<cdiv(ETOT, TB), TB, 0, stream>>>(ei, esb, edb, mb);
        edge_sum_kernel<<<cdiv(ETOT, TB), TB, 0, stream>>>(ei, esb, edb, mb, sb);
        edge_alpha_kernel<<<cdiv(ETOT * 4, TB), TB, 0, stream>>>(ei, esb, edb, mb, sb, alpha);

        fill_kernel<<<cdiv(NN * DOUT, TB), TB, 0, stream>>>(agg, 0.f, NN * DOUT);
        if (l == 2)
            edge_agg_kernel<32, 128><<<cdiv(ETOT * 32, TB), TB, 0, stream>>>(ei, hW, alpha, agg);
        else
            edge_agg_kernel<64, 256><<<cdiv(ETOT * 64, TB), TB, 0, stream>>>(ei, hW, alpha, agg);

        fill_kernel<<<1, 128, 0, stream>>>(stat, 0.f, 128);
        const int RPB = 128;
        if (l == 2) {
            bn_stats_kernel<32><<<cdiv(NN, RPB), 32, 0, stream>>>(agg, BI[l], stat, stat + 64, RPB);
            bn_norm_kernel<32, false><<<cdiv(NN * 32, TB), TB, 0, stream>>>(
                agg, BI[l], GA[l], BE[l], stat, stat + 64, hOut);
        } else {
            bn_stats_kernel<64><<<cdiv(NN, RPB), 64, 0, stream>>>(agg, BI[l], stat, stat + 64, RPB);
            bn_norm_kernel<64, true><<<cdiv(NN * 64, TB), TB, 0, stream>>>(
                agg, BI[l], GA[l], BE[l], stat, stat + 64, hA);
        }
    }

    // pooling
    fill_kernel<<<1, 64, 0, stream>>>(cnts, 0.f, 64);
    fill_kernel<<<cdiv(GG * 32, TB), TB, 0, stream>>>(gsum, 0.f, GG * 32);
    fill_kernel<<<cdiv(GG * 32, TB), TB, 0, stream>>>(gmaxb, NEG_INF, GG * 32);
    pool_kernel<<<cdiv(NN * 32, TB), TB, 0, stream>>>(hOut, batch, cnts, gsum, gmaxb);
    gemb_kernel<<<cdiv(GG * 32, TB), TB, 0, stream>>>(cnts, gsum, gmaxb, gembW, gembOut);
    heads_kernel<<<1, 128, 0, stream>>>(gembW, ew1, eb1, ew2, eb2, mw1, mb1, mw2, mb2,
                                        ethOut, manOut);
}